// PointNetSetAbstraction_59820304499057
// MI455X (gfx1250) — compile-verified
//
#include <hip/hip_runtime.h>
#include <hip/hip_bf16.h>

typedef __attribute__((ext_vector_type(2))) float v2f;
typedef __attribute__((ext_vector_type(8))) float v8f;
typedef __attribute__((ext_vector_type(4))) unsigned int u32x4;
typedef __attribute__((ext_vector_type(8))) int i32x8;
typedef __attribute__((ext_vector_type(4))) int i32x4;

#define BATCH   32
#define NPTS    65536
#define NSAMP   2048
#define PTOT    (BATCH * NSAMP)   // 65536 total sampled points
#define BN_EPS  1e-5f
#define NPW     4                 // N-tiles per wave in the WMMA GEMM

// ---------------------------------------------------------------------------
// 1) Farthest point sampling: one workgroup per batch, 1024 threads (32 waves).
//    dist[] lives in a global scratch slab (L2-resident, 8 MB total).
// ---------------------------------------------------------------------------
__global__ __launch_bounds__(1024)
void fps_kernel(const float* __restrict__ xyz, float* __restrict__ dist,
                int* __restrict__ idx) {
    const int b = blockIdx.x;
    const int t = threadIdx.x;                 // 0..1023
    const float* xb = xyz + (size_t)b * NPTS * 3;
    float* db = dist + (size_t)b * NPTS;

    __shared__ float cen[3];
    __shared__ float rv[1024];
    __shared__ int   ri[1024];
    __shared__ int   farS;

    for (int p = t; p < NPTS; p += 1024) db[p] = 1e10f;
    if (t == 0) farS = 0;
    __syncthreads();

    for (int it = 0; it < NSAMP; ++it) {
        const int far = farS;
        if (t == 0) {
            idx[b * NSAMP + it] = far;               // scan emits *old* far
            cen[0] = xb[far * 3 + 0];
            cen[1] = xb[far * 3 + 1];
            cen[2] = xb[far * 3 + 2];
        }
        __syncthreads();
        const float cx = cen[0], cy = cen[1], cz = cen[2];

        float bm = -1.0f;
        int   bi = 0;
        for (int p = t; p < NPTS; p += 1024) {       // 64 points/thread
            if (p + 1024 < NPTS)                      // hide L2 latency
                __builtin_prefetch(&xb[(p + 1024) * 3], 0, 0);
            const float dx = xb[p * 3 + 0] - cx;
            const float dy = xb[p * 3 + 1] - cy;
            const float dz = xb[p * 3 + 2] - cz;
            const float d  = dx * dx + dy * dy + dz * dz;
            const float nd = fminf(db[p], d);
            db[p] = nd;
            if (nd > bm) { bm = nd; bi = p; }        // strict > keeps lowest p
        }
        rv[t] = bm; ri[t] = bi;
        __syncthreads();
        for (int s = 512; s > 0; s >>= 1) {          // argmax, first-max ties
            if (t < s) {
                const float v2 = rv[t + s];
                const int   i2 = ri[t + s];
                if (v2 > rv[t] || (v2 == rv[t] && i2 < ri[t])) {
                    rv[t] = v2; ri[t] = i2;
                }
            }
            __syncthreads();
        }
        if (t == 0) farS = ri[0];
        __syncthreads();
    }
}

// ---------------------------------------------------------------------------
// 2) Gather: new_xyz to d_out ([B,S,3]) and channel-major xg ([3, PTOT]).
// ---------------------------------------------------------------------------
__global__ void gather_kernel(const float* __restrict__ xyz,
                              const int* __restrict__ idx,
                              float* __restrict__ new_xyz,
                              float* __restrict__ xg) {
    const int i = blockIdx.x * blockDim.x + threadIdx.x;   // i = b*NSAMP + s
    if (i >= PTOT) return;
    const int b = i >> 11;                                  // / NSAMP
    const int p = idx[i];
    const float* src = xyz + ((size_t)b * NPTS + p) * 3;
    const float x0 = src[0], x1 = src[1], x2 = src[2];
    new_xyz[i * 3 + 0] = x0;
    new_xyz[i * 3 + 1] = x1;
    new_xyz[i * 3 + 2] = x2;
    xg[0 * PTOT + i] = x0;
    xg[1 * PTOT + i] = x1;
    xg[2 * PTOT + i] = x2;
}

// ---------------------------------------------------------------------------
// 3) Pointwise conv as fp32 WMMA GEMM: y[o,p] = sum_c w[o,c]*x[c,p] + bias[o].
//    grid.y = Cout/16 (mtile), grid.x covers P/16 n-tiles, 8 waves * NPW each.
//    The 16xCin weight tile (shared by all waves of the block) is staged into
//    LDS by the Tensor Data Mover (tensor_load_to_lds + s_wait_tensorcnt).
//    WMMA fp32 layouts (wave32, ISA 7.12.2):
//      A 16x4 : lane<16 -> M=lane, K={k0,k0+1}; lane>=16 -> M=lane-16, K={k0+2,k0+3}
//      B 4x16 : lane<16 -> N=lane, rows {k0,k0+1}; lane>=16 -> rows {k0+2,k0+3}
//      C/D    : vgpr r -> row r + 8*(lane/16), col lane%16
// ---------------------------------------------------------------------------
__global__ __launch_bounds__(256)
void conv_wmma_kernel(const float* __restrict__ x, const float* __restrict__ w,
                      const float* __restrict__ bias, float* __restrict__ y,
                      int Cin, int Cout, int P) {
    const int mtile  = blockIdx.y;
    const int waveId = threadIdx.x >> 5;
    const int lane   = threadIdx.x & 31;
    const int half   = lane >> 4;
    const int ml     = lane & 15;
    const int ntile0 = (blockIdx.x * 8 + waveId) * NPW;

    __shared__ float wtile[16 * 128];          // max 16 x Cin fp32 (8 KB)

    // --- TDM: DMA the 16 x Cin weight tile into LDS (one wave issues) -----
    if (threadIdx.x == 0) {
        const unsigned long long ga =
            (unsigned long long)(uintptr_t)(w + (size_t)mtile * 16 * Cin);
        const unsigned lds = (unsigned)(uintptr_t)&wtile[0];
        u32x4 g0;
        g0[0] = 1u;                                   // count=1 (valid D#)
        g0[1] = lds;                                  // lds_addr
        g0[2] = (unsigned)(ga & 0xFFFFFFFFu);         // global_addr[31:0]
        g0[3] = (unsigned)((ga >> 32) & 0x01FFFFFFu)  // global_addr[56:32]
              | 0x80000000u;                          // type=2 ("image")
        i32x8 g1;
        g1[0] = 0x00020000;                           // data_size=2 (4 bytes)
        g1[1] = (Cin & 0xFFFF) << 16;                 // tensor_dim0 lo16
        g1[2] = ((Cin >> 16) & 0xFFFF) | (16 << 16);  // td0 hi16 | tensor_dim1=16
        g1[3] = (Cin & 0xFFFF) << 16;                 // td1 hi16=0 | tile_dim0=Cin
        g1[4] = 16;                                   // tile_dim1=16, tile_dim2=0
        g1[5] = Cin;                                  // tensor_dim0_stride lo32
        g1[6] = 0;
        g1[7] = 0;
        i32x4 gz4 = {0, 0, 0, 0};
        i32x8 gz8 = {0, 0, 0, 0, 0, 0, 0, 0};
        __builtin_amdgcn_tensor_load_to_lds(g0, g1, gz4, gz4, gz8, 0);
        __builtin_amdgcn_s_wait_tensorcnt(0);
    }
    __syncthreads();

    v8f acc[NPW];
#pragma unroll
    for (int j = 0; j < NPW; ++j) acc[j] = (v8f){};

    for (int k0 = 0; k0 < Cin; k0 += 4) {
        const int ka = k0 + 2 * half;
        v2f a;
        a.x = (ka     < Cin) ? wtile[ml * Cin + ka]     : 0.0f;
        a.y = (ka + 1 < Cin) ? wtile[ml * Cin + ka + 1] : 0.0f;
#pragma unroll
        for (int j = 0; j < NPW; ++j) {
            const int n = (ntile0 + j) * 16 + ml;
            v2f bv;
            bv.x = (ka     < Cin) ? x[(size_t)(ka)     * P + n] : 0.0f;
            bv.y = (ka + 1 < Cin) ? x[(size_t)(ka + 1) * P + n] : 0.0f;
            acc[j] = __builtin_amdgcn_wmma_f32_16x16x4_f32(
                         /*neg_a=*/false, a, /*neg_b=*/false, bv,
                         /*c_mod=*/(short)0, acc[j],
                         /*reuse_a=*/false, /*reuse_b=*/false);
        }
    }
#pragma unroll
    for (int j = 0; j < NPW; ++j) {
#pragma unroll
        for (int r = 0; r < 8; ++r) {
            const int o = mtile * 16 + r + 8 * half;
            y[(size_t)o * P + (ntile0 + j) * 16 + ml] = acc[j][r] + bias[o];
        }
    }
}

// ---------------------------------------------------------------------------
// 4) BN batch statistics: one workgroup per output channel.
//    stats[2*o] = mean, stats[2*o+1] = rsqrt(var + eps)
// ---------------------------------------------------------------------------
__global__ __launch_bounds__(256)
void bn_stats_kernel(const float* __restrict__ y, float* __restrict__ stats, int P) {
    const int o = blockIdx.x;
    const int t = threadIdx.x;
    float s1 = 0.0f, s2 = 0.0f;
    for (int p = t; p < P; p += 256) {
        const float v = y[(size_t)o * P + p];
        s1 += v; s2 += v * v;
    }
    __shared__ float a1[256], a2[256];
    a1[t] = s1; a2[t] = s2;
    __syncthreads();
    for (int s = 128; s > 0; s >>= 1) {
        if (t < s) { a1[t] += a1[t + s]; a2[t] += a2[t + s]; }
        __syncthreads();
    }
    if (t == 0) {
        const float inv_n = 1.0f / (float)P;
        const float mu  = a1[0] * inv_n;
        const float var = a2[0] * inv_n - mu * mu;   // biased variance
        stats[2 * o + 0] = mu;
        stats[2 * o + 1] = rsqrtf(var + BN_EPS);
    }
}

// ---------------------------------------------------------------------------
// 5) BN apply + ReLU, in place over [Cout, P]. grid = (P/256, Cout).
// ---------------------------------------------------------------------------
__global__ __launch_bounds__(256)
void bn_apply_kernel(float* __restrict__ y, const float* __restrict__ stats,
                     const float* __restrict__ g, const float* __restrict__ be,
                     int P) {
    const int o = blockIdx.y;
    const size_t i = (size_t)o * P + (size_t)blockIdx.x * 256 + threadIdx.x;
    const float mu  = stats[2 * o + 0];
    const float inv = stats[2 * o + 1];
    const float v = (y[i] - mu) * inv * g[o] + be[o];
    y[i] = fmaxf(v, 0.0f);
}

// ---------------------------------------------------------------------------
// 6) Max-pool over S per (batch, channel): grid = (Cout, BATCH).
//    y layout [O, P] with p = b*NSAMP + s  ->  contiguous 2048-run.
// ---------------------------------------------------------------------------
__global__ __launch_bounds__(256)
void maxpool_kernel(const float* __restrict__ y, float* __restrict__ feat,
                    int Cout, int P) {
    const int o = blockIdx.x;
    const int b = blockIdx.y;
    const int t = threadIdx.x;
    const float* row = y + (size_t)o * P + (size_t)b * NSAMP;
    float m = -3.4e38f;
    for (int s = t; s < NSAMP; s += 256) m = fmaxf(m, row[s]);
    __shared__ float red[256];
    red[t] = m;
    __syncthreads();
    for (int s = 128; s > 0; s >>= 1) {
        if (t < s) red[t] = fmaxf(red[t], red[t + s]);
        __syncthreads();
    }
    if (t == 0) feat[b * Cout + o] = red[0];
}

// ---------------------------------------------------------------------------
// Host launcher
// ---------------------------------------------------------------------------
extern "C" void kernel_launch(void* const* d_in, const int* in_sizes, int n_in,
                              void* d_out, int out_size, void* d_ws, size_t ws_size,
                              hipStream_t stream) {
    const float* xyz = (const float*)d_in[0];
    const float* w0 = (const float*)d_in[1];
    const float* b0 = (const float*)d_in[2];
    const float* g0 = (const float*)d_in[3];
    const float* be0 = (const float*)d_in[4];
    const float* w1 = (const float*)d_in[5];
    const float* b1 = (const float*)d_in[6];
    const float* g1 = (const float*)d_in[7];
    const float* be1 = (const float*)d_in[8];
    const float* w2 = (const float*)d_in[9];
    const float* b2 = (const float*)d_in[10];
    const float* g2 = (const float*)d_in[11];
    const float* be2 = (const float*)d_in[12];

    float* out = (float*)d_out;
    float* new_xyz  = out;                              // 196608 floats
    float* features = out + (size_t)BATCH * NSAMP * 3;  // BATCH*256

    // Workspace layout (bytes)
    char* ws = (char*)d_ws;
    size_t off = 0;
    float* dist = (float*)(ws + off); off += (size_t)BATCH * NPTS * sizeof(float);      // 8 MB
    int*   idx  = (int*)  (ws + off); off += (size_t)BATCH * NSAMP * sizeof(int);       // 256 KB
    float* xg   = (float*)(ws + off); off += (size_t)3 * PTOT * sizeof(float);          // 768 KB
    float* stats= (float*)(ws + off); off += (size_t)2 * 256 * sizeof(float);           // 2 KB
    float* bufA = (float*)(ws + off); off += (size_t)256 * PTOT * sizeof(float);        // 64 MB (y0, y2)
    float* bufB = (float*)(ws + off); off += (size_t)128 * PTOT * sizeof(float);        // 32 MB (y1)
    (void)off; (void)ws_size; (void)in_sizes; (void)n_in; (void)out_size;

    // 1) FPS
    fps_kernel<<<BATCH, 1024, 0, stream>>>(xyz, dist, idx);

    // 2) Gather
    gather_kernel<<<PTOT / 256, 256, 0, stream>>>(xyz, idx, new_xyz, xg);

    const int NTILE_BLOCKS = (PTOT / 16) / (8 * NPW);   // 128

    // 3) conv0 (3 -> 64) + BN + ReLU
    {
        const int Cin = 3, Cout = 64, P = PTOT;
        conv_wmma_kernel<<<dim3(NTILE_BLOCKS, Cout / 16), 256, 0, stream>>>(
            xg, w0, b0, bufA, Cin, Cout, P);
        bn_stats_kernel<<<Cout, 256, 0, stream>>>(bufA, stats, P);
        bn_apply_kernel<<<dim3(P / 256, Cout), 256, 0, stream>>>(bufA, stats, g0, be0, P);
    }
    // 4) conv1 (64 -> 128) + BN + ReLU
    {
        const int Cin = 64, Cout = 128, P = PTOT;
        conv_wmma_kernel<<<dim3(NTILE_BLOCKS, Cout / 16), 256, 0, stream>>>(
            bufA, w1, b1, bufB, Cin, Cout, P);
        bn_stats_kernel<<<Cout, 256, 0, stream>>>(bufB, stats, P);
        bn_apply_kernel<<<dim3(P / 256, Cout), 256, 0, stream>>>(bufB, stats, g1, be1, P);
    }
    // 5) conv2 (128 -> 256) + BN + ReLU
    {
        const int Cin = 128, Cout = 256, P = PTOT;
        conv_wmma_kernel<<<dim3(NTILE_BLOCKS, Cout / 16), 256, 0, stream>>>(
            bufB, w2, b2, bufA, Cin, Cout, P);
        bn_stats_kernel<<<Cout, 256, 0, stream>>>(bufA, stats, P);
        bn_apply_kernel<<<dim3(P / 256, Cout), 256, 0, stream>>>(bufA, stats, g2, be2, P);
    }

    // 6) Global max-pool -> features
    maxpool_kernel<<<dim3(256, BATCH), 256, 0, stream>>>(bufA, features, 256, PTOT);
}